// Codebook_51049981280413
// MI455X (gfx1250) — compile-verified
//
#include <hip/hip_runtime.h>

typedef float v2f __attribute__((ext_vector_type(2)));
typedef float v8f __attribute__((ext_vector_type(8)));

#define D_IN     512
#define D_E      256
#define N_CODE   1024
#define T_LEN    4096
#define N_BATCH  16
#define BT       (N_BATCH * T_LEN)        /* 65536 tokens */
#define ZOUT_ELEMS (BT * D_IN)            /* 33554432 */
#define LOSS_OFF   ZOUT_ELEMS
#define CODE_OFF   (ZOUT_ELEMS + 32)
#define WAVES_PB   2
#define N_TILES    (BT / 16)              /* 4096 wave tiles */
#define TILES_PER_BATCH (N_TILES / N_BATCH) /* 256 */

__device__ __forceinline__ v8f wmma_f32(v2f a, v2f b, v8f c) {
  // D = A(16x4,f32) * B(4x16,f32) + C(16x16,f32); 8-arg VOP3P form
  return __builtin_amdgcn_wmma_f32_16x16x4_f32(false, a, false, b,
                                               (short)0, c, false, false);
}

// ---------- prep: codebook squared norms ----------
__global__ void vq_prep(const float* __restrict__ cb, float* __restrict__ cnorm) {
  int i = blockIdx.x * blockDim.x + threadIdx.x;
  if (i < N_CODE) {
    const float* row = cb + (size_t)i * D_E;
    float s = 0.f;
    for (int j = 0; j < D_E; j += 4) {
      float4 v = *(const float4*)(row + j);
      s += v.x * v.x + v.y * v.y + v.z * v.z + v.w * v.w;
    }
    cnorm[i] = s;
  }
}

// ---------- fused VQ kernel: 1 wave == 16 tokens ----------
__global__ __launch_bounds__(64) void vq_fused(
    const float* __restrict__ z, const float* __restrict__ cb,
    const float* __restrict__ wdn, const float* __restrict__ wup,
    const float* __restrict__ cnorm, float* __restrict__ wave_loss,
    float* __restrict__ zout, float* __restrict__ code_out)
{
  __shared__ float ze[WAVES_PB][16][260];   // z_e then z_q; pad 260 vs banks
  __shared__ float dtile[WAVES_PB][16][17]; // distance staging
  __shared__ int   codes[WAVES_PB][16];

  const int tid  = threadIdx.x;
  const int wv   = tid >> 5;
  const int lane = tid & 31;
  const int lo   = lane & 15;   // column/row-in-tile index
  const int hi   = lane >> 4;   // K sub-pair select (0 or 1)
  const int tile = blockIdx.x * WAVES_PB + wv;
  const int tok0 = tile * 16;

  const v8f vzero = {0.f, 0.f, 0.f, 0.f, 0.f, 0.f, 0.f, 0.f};

  // ===== Stage A: z_e[16,256] = z_tile[16,512] @ w_down^T =====
  v8f acc[16];
#pragma unroll
  for (int nt = 0; nt < 16; ++nt) acc[nt] = vzero;

  const float* za = z   + (size_t)(tok0 + lo) * D_IN + 2 * hi; // A: row tok0+lo
  const float* wb = wdn + (size_t)lo * D_IN + 2 * hi;          // B: w_down[n][k]

  for (int k = 0; k < D_IN; k += 4) {
    v2f a = *(const v2f*)(za + k);
#pragma unroll
    for (int nt = 0; nt < 16; ++nt) {
      v2f b = *(const v2f*)(wb + (size_t)nt * 16 * D_IN + k);
      acc[nt] = wmma_f32(a, b, acc[nt]);
    }
  }
  // park z_e in LDS (C layout: vgpr v -> row v + 8*hi, col nt*16+lo)
#pragma unroll
  for (int nt = 0; nt < 16; ++nt)
#pragma unroll
    for (int v = 0; v < 8; ++v)
      ze[wv][v + 8 * hi][nt * 16 + lo] = acc[nt][v];
  asm volatile("s_wait_dscnt 0x0" ::: "memory");

  // ===== Stage B: distances to 1024 codes, running argmin =====
  float best = 3.4e38f;
  int bestIdx = 0;
  const float* zep = &ze[wv][lo][2 * hi];
  for (int c0 = 0; c0 < N_CODE; c0 += 16) {
    if (c0 + 16 < N_CODE)
      __builtin_prefetch(cb + (size_t)(c0 + 16 + lo) * D_E, 0, 0);
    v8f d = vzero;
    const float* cbp = cb + (size_t)(c0 + lo) * D_E + 2 * hi;
    for (int k = 0; k < D_E; k += 4) {
      v2f a = *(const v2f*)(zep + k);
      v2f b = *(const v2f*)(cbp + k);
      d = wmma_f32(a, b, d);
    }
    float cn = cnorm[c0 + lo];
#pragma unroll
    for (int v = 0; v < 8; ++v)
      dtile[wv][v + 8 * hi][lo] = cn - 2.0f * d[v]; // ||z_e||^2 const per row
    asm volatile("s_wait_dscnt 0x0" ::: "memory");  // wave-lockstep LDS sync
    if (lane < 16) {
#pragma unroll
      for (int j = 0; j < 16; ++j) {
        float dd = dtile[wv][lane][j];
        if (dd < best) { best = dd; bestIdx = c0 + j; }
      }
    }
    asm volatile("s_wait_dscnt 0x0" ::: "memory");
  }
  if (lane < 16) {
    codes[wv][lane] = bestIdx;
    code_out[tok0 + lane] = (float)bestIdx;
  }
  asm volatile("s_wait_dscnt 0x0" ::: "memory");

  // ===== Loss + gather: overwrite z_e in LDS with z_q =====
  float lsum = 0.f;
  for (int e = lane; e < 16 * D_E; e += 32) {
    int m = e >> 8;
    int n = e & (D_E - 1);
    float zq  = cb[(size_t)codes[wv][m] * D_E + n];
    float zev = ze[wv][m][n];
    float df  = zq - zev;
    lsum += df * df;
    ze[wv][m][n] = zq;
  }
  asm volatile("s_wait_dscnt 0x0" ::: "memory");
#pragma unroll
  for (int off = 16; off > 0; off >>= 1)
    lsum += __shfl_down(lsum, off, 32);
  if (lane == 0) wave_loss[tile] = lsum;

  // ===== Stage D: z_out[16,512] = z_q[16,256] @ w_up^T =====
  for (int nt = 0; nt < 32; ++nt) {
    v8f o = vzero;
    const float* wbp = wup + (size_t)(nt * 16 + lo) * D_E + 2 * hi;
    for (int k = 0; k < D_E; k += 4) {
      v2f a = *(const v2f*)(zep + k);
      v2f b = *(const v2f*)(wbp + k);
      o = wmma_f32(a, b, o);
    }
#pragma unroll
    for (int v = 0; v < 8; ++v)
      zout[(size_t)(tok0 + v + 8 * hi) * D_IN + nt * 16 + lo] = o[v];
  }
}

// ---------- deterministic per-batch loss reduction ----------
__global__ void vq_finalize(const float* __restrict__ wave_loss,
                            float* __restrict__ loss_out) {
  __shared__ float s[TILES_PER_BATCH];
  int b = blockIdx.x, t = threadIdx.x;
  s[t] = wave_loss[b * TILES_PER_BATCH + t];
  __syncthreads();
  for (int off = TILES_PER_BATCH / 2; off > 0; off >>= 1) {
    if (t < off) s[t] += s[t + off];
    __syncthreads();
  }
  if (t == 0) {
    float v = s[0] * (1.0f / ((float)T_LEN * (float)D_E));
    loss_out[b]            = v;  // commitment loss
    loss_out[N_BATCH + b]  = v;  // codebook loss (same forward value)
  }
}

extern "C" void kernel_launch(void* const* d_in, const int* in_sizes, int n_in,
                              void* d_out, int out_size, void* d_ws, size_t ws_size,
                              hipStream_t stream) {
  (void)in_sizes; (void)n_in; (void)out_size; (void)ws_size;
  const float* z   = (const float*)d_in[0];
  const float* cb  = (const float*)d_in[1];
  const float* wdn = (const float*)d_in[2];
  const float* wup = (const float*)d_in[3];
  float* out   = (float*)d_out;
  float* cnorm = (float*)d_ws;             // 1024 floats
  float* wloss = cnorm + N_CODE;           // 4096 floats

  vq_prep<<<(N_CODE + 255) / 256, 256, 0, stream>>>(cb, cnorm);
  vq_fused<<<BT / (WAVES_PB * 16), WAVES_PB * 32, 0, stream>>>(
      z, cb, wdn, wup, cnorm, wloss, out, out + CODE_OFF);
  vq_finalize<<<N_BATCH, TILES_PER_BATCH, 0, stream>>>(wloss, out + LOSS_OFF);
}